// MoEConv_29755533427437
// MI455X (gfx1250) — compile-verified
//
#include <hip/hip_runtime.h>
#include <hip/hip_bf16.h>
#include <math.h>

typedef __attribute__((ext_vector_type(2))) float v2f;
typedef __attribute__((ext_vector_type(8))) float v8f;

#define IN_C 64
#define OUT_C 64
#define NK 8
#define HID 128

// ---------------------------------------------------------------------------
// CDNA5 device-scope f32 atomic max (GLOBAL_ATOMIC_MAX_NUM_F32, no return).
// ---------------------------------------------------------------------------
__device__ __forceinline__ void atomic_max_f32(float* addr, float val) {
  asm volatile("global_atomic_max_num_f32 %0, %1, off scope:SCOPE_DEV"
               :
               : "v"(addr), "v"(val)
               : "memory");
}

// ---------------------------------------------------------------------------
// Kernel 1: Y[n, k*64+o] = sum_i X[n,i] * EW[k,i,o]   (50000x64 @ 64x512)
// One wave per 16x16 output tile, K=64 via 16x V_WMMA_F32_16X16X4_F32.
// A layout (16x4 f32): lane l holds A[m=l%16][k0], A[m][k0+1], k0=(l/16)*2.
// B layout (4x16 f32): lane l holds B[k0][n=l%16], B[k0+1][n].
// D layout (16x16 f32): VGPR r, lane l -> D[r + 8*(l/16)][l%16].
// ---------------------------------------------------------------------------
__global__ void expert_gemm_kernel(const float* __restrict__ X,
                                   const float* __restrict__ EW,
                                   float* __restrict__ Y, int N) {
  const int lane  = threadIdx.x & 31;
  const int wave  = threadIdx.x >> 5;
  const int row0  = blockIdx.x * 16;
  const int ctile = blockIdx.y * 8 + wave;      // 0..31 (512 output cols)
  const int n     = lane & 15;
  const int kh    = lane >> 4;                  // 0 or 1

  const int colbase = ctile * 16;
  const int expert  = colbase >> 6;             // which expert's 64-col slab
  const float* Bp   = EW + (size_t)expert * (IN_C * OUT_C) + (colbase & 63) + n;

  const int arow  = (row0 + n < N) ? (row0 + n) : (N - 1);
  const float* Ap = X + (size_t)arow * IN_C;

  v8f acc = {};
#pragma unroll
  for (int kk = 0; kk < IN_C; kk += 4) {
    const int k0 = kk + kh * 2;                 // even -> 8B-aligned float2
    const float2 av = *(const float2*)(Ap + k0);
    v2f a, b;
    a.x = av.x;
    a.y = av.y;
    b.x = Bp[(size_t)k0 * OUT_C];
    b.y = Bp[(size_t)(k0 + 1) * OUT_C];
    acc = __builtin_amdgcn_wmma_f32_16x16x4_f32(false, a, false, b,
                                                (short)0, acc, false, false);
  }
#pragma unroll
  for (int r = 0; r < 8; ++r) {
    const int row = row0 + r + 8 * kh;
    if (row < N) Y[(size_t)row * (NK * OUT_C) + colbase + n] = acc[r];
  }
}

// ---------------------------------------------------------------------------
// Kernel 2: fill the segment-max accumulator with -inf.
// ---------------------------------------------------------------------------
__global__ void init_neginf_kernel(float* __restrict__ p, int total) {
  const int i = blockIdx.x * blockDim.x + threadIdx.x;
  if (i < total) p[i] = -__builtin_inff();
}

// ---------------------------------------------------------------------------
// Kernel 3: one wave per edge. Gate (top-2 softmax over 8 logits from the
// 2-D position diff), gather two 64-float expert rows of Y, blend, and
// atomic-max scatter into out[dst]. Lanes cover 64 channels as float2.
// ---------------------------------------------------------------------------
__global__ void edge_kernel(const float* __restrict__ pos,
                            const long long* __restrict__ ei,
                            const float* __restrict__ gW,   // (2,8)
                            const float* __restrict__ gb,   // (8)
                            const float* __restrict__ Y,    // (N,512)
                            float* __restrict__ outmax,     // (N,64)
                            int E) {
  const int lane = threadIdx.x & 31;
  const int wave = threadIdx.x >> 5;
  const int e    = blockIdx.x * 8 + wave;
  if (e >= E) return;

  const long long dst = ei[2 * (long long)e];
  const long long src = ei[2 * (long long)e + 1];

  const float dp0 = pos[src * 2]     - pos[dst * 2];
  const float dp1 = pos[src * 2 + 1] - pos[dst * 2 + 1];

  float lg[NK];
#pragma unroll
  for (int k = 0; k < NK; ++k)
    lg[k] = fmaf(dp0, gW[k], fmaf(dp1, gW[NK + k], gb[k]));

  // top-2 with lowest-index-first tie breaking (matches jax.lax.top_k)
  int k1 = 0;
  float v1 = lg[0];
#pragma unroll
  for (int k = 1; k < NK; ++k)
    if (lg[k] > v1) { v1 = lg[k]; k1 = k; }
  int k2 = -1;
  float v2 = -__builtin_inff();
#pragma unroll
  for (int k = 0; k < NK; ++k)
    if (k != k1 && lg[k] > v2) { v2 = lg[k]; k2 = k; }

  const float t  = __expf(v2 - v1);          // <= 1, numerically stable
  const float iv = 1.0f / (1.0f + t);
  const float w1 = iv;
  const float w2 = t * iv;

  const float2 m1 = ((const float2*)(Y + (size_t)src * (NK * OUT_C) + k1 * OUT_C))[lane];
  const float2 m2 = ((const float2*)(Y + (size_t)src * (NK * OUT_C) + k2 * OUT_C))[lane];
  const float r0 = fmaf(w1, m1.x, w2 * m2.x);
  const float r1 = fmaf(w1, m1.y, w2 * m2.y);

  float* o = outmax + (size_t)dst * OUT_C + lane * 2;
  atomic_max_f32(o, r0);
  atomic_max_f32(o + 1, r1);
}

// ---------------------------------------------------------------------------
// Kernel 4: replace -inf (isolated node) with 0, in place.
// ---------------------------------------------------------------------------
__global__ void fixup_kernel(float* __restrict__ p, int total) {
  const int i = blockIdx.x * blockDim.x + threadIdx.x;
  if (i < total) {
    const float v = p[i];
    p[i] = (v == -__builtin_inff()) ? 0.0f : v;
  }
}

// ---------------------------------------------------------------------------
// Kernel 5: h = relu(out @ W1) @ W2 + out, all WMMA f32.
// Block = 256 threads (8 waves), 16 rows per block.
// Stage 1: wave w computes 16x16 tile of relu(out@W1) cols [16w,16w+16),
//          stored to LDS (so stage 2 can re-read it in WMMA A layout).
// Stage 2: waves 0..3 compute the 16x16 output tiles over K=128, add residual.
// ---------------------------------------------------------------------------
__global__ void mlp_kernel(const float* __restrict__ out,  // (N,64) fixed-up
                           const float* __restrict__ W1,   // (64,128)
                           const float* __restrict__ W2,   // (128,64)
                           float* __restrict__ h,          // (N,64) result
                           int N) {
  __shared__ float T[16 * HID];

  const int lane = threadIdx.x & 31;
  const int wave = threadIdx.x >> 5;
  const int row0 = blockIdx.x * 16;
  const int n    = lane & 15;
  const int kh   = lane >> 4;

  // ---- stage 1: acc = out[row0:row0+16, :] @ W1[:, 16w:16w+16] ----
  {
    const int arow  = (row0 + n < N) ? (row0 + n) : (N - 1);
    const float* Ap = out + (size_t)arow * OUT_C;
    const int cb    = wave * 16;
    v8f acc = {};
#pragma unroll
    for (int kk = 0; kk < OUT_C; kk += 4) {
      const int k0 = kk + kh * 2;
      const float2 av = *(const float2*)(Ap + k0);
      v2f a, b;
      a.x = av.x;
      a.y = av.y;
      b.x = W1[(size_t)k0 * HID + cb + n];
      b.y = W1[(size_t)(k0 + 1) * HID + cb + n];
      acc = __builtin_amdgcn_wmma_f32_16x16x4_f32(false, a, false, b,
                                                  (short)0, acc, false, false);
    }
#pragma unroll
    for (int r = 0; r < 8; ++r)
      T[(r + 8 * kh) * HID + cb + n] = fmaxf(acc[r], 0.0f);
  }
  __syncthreads();

  // ---- stage 2: waves 0..3 -> D = T @ W2[:, 16w:16w+16] + residual ----
  if (wave < 4) {
    const int cb = wave * 16;
    v8f acc = {};
#pragma unroll
    for (int kk = 0; kk < HID; kk += 4) {
      const int k0 = kk + kh * 2;
      v2f a, b;
      a.x = T[n * HID + k0];          // A[m=n][k0]
      a.y = T[n * HID + k0 + 1];
      b.x = W2[(size_t)k0 * OUT_C + cb + n];
      b.y = W2[(size_t)(k0 + 1) * OUT_C + cb + n];
      acc = __builtin_amdgcn_wmma_f32_16x16x4_f32(false, a, false, b,
                                                  (short)0, acc, false, false);
    }
#pragma unroll
    for (int r = 0; r < 8; ++r) {
      const int row = row0 + r + 8 * kh;
      if (row < N) {
        const size_t idx = (size_t)row * OUT_C + cb + n;
        h[idx] = acc[r] + out[idx];
      }
    }
  }
}

// ---------------------------------------------------------------------------
// Launch: inputs in setup_inputs() order:
//   0:x (N,64) f32   1:pos (N,2) f32   2:edge_index (E,2) i64
//   3:expert_weights (8,64,64) f32     4:gate_W (2,8)     5:gate_b (8)
//   6:W1 (64,128)    7:W2 (128,64)
// Workspace: Y (N*512 f32) | outmax (N*64 f32)
// ---------------------------------------------------------------------------
extern "C" void kernel_launch(void* const* d_in, const int* in_sizes, int n_in,
                              void* d_out, int out_size, void* d_ws, size_t ws_size,
                              hipStream_t stream) {
  const float*     x   = (const float*)d_in[0];
  const float*     pos = (const float*)d_in[1];
  const long long* ei  = (const long long*)d_in[2];
  const float*     ew  = (const float*)d_in[3];
  const float*     gW  = (const float*)d_in[4];
  const float*     gb  = (const float*)d_in[5];
  const float*     W1  = (const float*)d_in[6];
  const float*     W2  = (const float*)d_in[7];
  float*           ho  = (float*)d_out;

  const int N = in_sizes[0] / IN_C;   // 50000
  const int E = in_sizes[2] / 2;      // 1600000

  float* Y      = (float*)d_ws;                               // N*512 f32
  float* outmax = (float*)((char*)d_ws + (size_t)N * NK * OUT_C * sizeof(float));

  const int mtiles = (N + 15) / 16;

  // 1) node-level expert GEMM (WMMA f32): Y = X @ concat_k(W_k)
  expert_gemm_kernel<<<dim3(mtiles, 4), 256, 0, stream>>>(x, ew, Y, N);

  // 2) accumulator = -inf
  {
    const int total = N * OUT_C;
    init_neginf_kernel<<<(total + 255) / 256, 256, 0, stream>>>(outmax, total);
  }

  // 3) gate + gather + atomic-max scatter (one wave per edge)
  edge_kernel<<<(E + 7) / 8, 256, 0, stream>>>(pos, ei, gW, gb, Y, outmax, E);

  // 4) -inf -> 0
  {
    const int total = N * OUT_C;
    fixup_kernel<<<(total + 255) / 256, 256, 0, stream>>>(outmax, total);
  }

  // 5) MLP + residual (WMMA f32)
  mlp_kernel<<<mtiles, 256, 0, stream>>>(outmax, W1, W2, ho, N);
}